// MOEFeedForward_26508538151527
// MI455X (gfx1250) — compile-verified
//
#include <hip/hip_runtime.h>
#include <hip/hip_bf16.h>
#include <math.h>

// Problem constants (from reference): B=4, S=2048 -> T=8192, H=1024, I=2752, E=8, TOPK=2
#define T_TOK 8192
#define H_DIM 1024
#define I_DIM 2752
#define E_NUM 8

typedef __attribute__((ext_vector_type(2))) float v2f;
typedef __attribute__((ext_vector_type(8))) float v8f;

#define LDSTR 36   // LDS row stride (floats): 144B -> 16B aligned, conflict-free

// ---------------------------------------------------------------------------
// Gate: one wave32 per token -> softmax over 8 experts -> top-2 -> renorm ->
// dense per-expert weight column w_et[e*T + t] (0 for unselected experts).
// ---------------------------------------------------------------------------
__global__ void moe_gate_kernel(const float* __restrict__ x,
                                const float* __restrict__ gw,
                                float* __restrict__ w_et) {
    const int wave = threadIdx.x >> 5;
    const int lane = threadIdx.x & 31;
    const int t = blockIdx.x * (blockDim.x >> 5) + wave;
    if (t >= T_TOK) return;

    const float* xr = x + (size_t)t * H_DIM;
    float logit[E_NUM];
#pragma unroll
    for (int e = 0; e < E_NUM; ++e) {
        const float* gr = gw + (size_t)e * H_DIM;
        float s = 0.f;
        for (int k = lane; k < H_DIM; k += 32) s += xr[k] * gr[k];
#pragma unroll
        for (int off = 16; off > 0; off >>= 1) s += __shfl_xor(s, off, 32);
        logit[e] = s;
    }
    if (lane == 0) {
        float m = logit[0];
#pragma unroll
        for (int e = 1; e < E_NUM; ++e) m = fmaxf(m, logit[e]);
        float p[E_NUM]; float den = 0.f;
#pragma unroll
        for (int e = 0; e < E_NUM; ++e) { p[e] = __expf(logit[e] - m); den += p[e]; }
        float inv = 1.f / den;
#pragma unroll
        for (int e = 0; e < E_NUM; ++e) p[e] *= inv;
        int i1 = 0;
#pragma unroll
        for (int e = 1; e < E_NUM; ++e) if (p[e] > p[i1]) i1 = e;
        int i2 = (i1 == 0) ? 1 : 0;
#pragma unroll
        for (int e = 0; e < E_NUM; ++e) if (e != i1 && p[e] > p[i2]) i2 = e;
        float rn = 1.f / (p[i1] + p[i2] + 1e-20f);
        float w1 = p[i1] * rn, w2 = p[i2] * rn;
#pragma unroll
        for (int e = 0; e < E_NUM; ++e)
            w_et[(size_t)e * T_TOK + t] = (e == i1) ? w1 : ((e == i2) ? w2 : 0.f);
    }
}

// ---------------------------------------------------------------------------
// GEMM1 (fused SwiGLU + routing weight):
//   hdn[mloc,n] = w[t] * silu(x·Wg^T) * (x·Wu^T)
// Block tile 256(M) x 64(N), 8 waves; each wave owns 32(M)x64(N) with dual
// (gate/up) accumulators -> each LDS B-fragment feeds 2 WMMAs.
// Double-buffered LDS K-slabs of 32.
// ---------------------------------------------------------------------------
__global__ void __launch_bounds__(256)
moe_gemm1_swiglu(const float* __restrict__ X,
                 const float* __restrict__ Wg,
                 const float* __restrict__ Wu,
                 const float* __restrict__ wrow,   // nullable (shared expert)
                 float* __restrict__ Hdn,
                 int mBase) {
    __shared__ float lX[2][256 * LDSTR];
    __shared__ float lG[2][64 * LDSTR];
    __shared__ float lU[2][64 * LDSTR];

    const int tid  = threadIdx.x;
    const int wave = tid >> 5;
    const int lane = tid & 31;
    const int lm   = lane & 15;
    const int kh   = (lane >> 4) << 1;       // K sub-offset 0/2 (f32 16x4 frag layout)

    const int nBase  = blockIdx.x * 64;
    const int mTile  = blockIdx.y * 256;
    const int mGlobT = mBase + mTile;

    const int ldRow = tid >> 3;              // 0..31 row step per iter chunk
    const int ldC4  = (tid & 7) << 2;        // 0,4,...,28

    v8f accG[2][4] = {}, accU[2][4] = {};

    // stage slab 0
    float4 rx[8], rg[2], ru[2];
#pragma unroll
    for (int it = 0; it < 8; ++it)
        rx[it] = *(const float4*)(X + (size_t)(mGlobT + ldRow + it * 32) * H_DIM + ldC4);
#pragma unroll
    for (int it = 0; it < 2; ++it) {
        rg[it] = *(const float4*)(Wg + (size_t)(nBase + ldRow + it * 32) * H_DIM + ldC4);
        ru[it] = *(const float4*)(Wu + (size_t)(nBase + ldRow + it * 32) * H_DIM + ldC4);
    }
#pragma unroll
    for (int it = 0; it < 8; ++it)
        *(float4*)(&lX[0][(ldRow + it * 32) * LDSTR + ldC4]) = rx[it];
#pragma unroll
    for (int it = 0; it < 2; ++it) {
        *(float4*)(&lG[0][(ldRow + it * 32) * LDSTR + ldC4]) = rg[it];
        *(float4*)(&lU[0][(ldRow + it * 32) * LDSTR + ldC4]) = ru[it];
    }
    __syncthreads();

    const int NS = H_DIM / 32;
    for (int s = 0; s < NS; ++s) {
        const int cur = s & 1;
        // prefetch next slab into registers while computing
        if (s + 1 < NS) {
            const int k0 = (s + 1) * 32;
#pragma unroll
            for (int it = 0; it < 8; ++it)
                rx[it] = *(const float4*)(X + (size_t)(mGlobT + ldRow + it * 32) * H_DIM + k0 + ldC4);
#pragma unroll
            for (int it = 0; it < 2; ++it) {
                rg[it] = *(const float4*)(Wg + (size_t)(nBase + ldRow + it * 32) * H_DIM + k0 + ldC4);
                ru[it] = *(const float4*)(Wu + (size_t)(nBase + ldRow + it * 32) * H_DIM + k0 + ldC4);
            }
        }
        const float* lXr0 = &lX[cur][(wave * 32 + lm) * LDSTR];
        const float* lXr1 = lXr0 + 16 * LDSTR;
#pragma unroll
        for (int kk = 0; kk < 32; kk += 4) {
            v2f a0, a1;
            a0.x = lXr0[kk + kh]; a0.y = lXr0[kk + kh + 1];
            a1.x = lXr1[kk + kh]; a1.y = lXr1[kk + kh + 1];
#pragma unroll
            for (int nt = 0; nt < 4; ++nt) {
                const float* gp = &lG[cur][(nt * 16 + lm) * LDSTR + kk + kh];
                v2f bg; bg.x = gp[0]; bg.y = gp[1];
                accG[0][nt] = __builtin_amdgcn_wmma_f32_16x16x4_f32(
                    false, a0, false, bg, (short)0, accG[0][nt], false, false);
                accG[1][nt] = __builtin_amdgcn_wmma_f32_16x16x4_f32(
                    false, a1, false, bg, (short)0, accG[1][nt], false, false);
                const float* up = &lU[cur][(nt * 16 + lm) * LDSTR + kk + kh];
                v2f bu; bu.x = up[0]; bu.y = up[1];
                accU[0][nt] = __builtin_amdgcn_wmma_f32_16x16x4_f32(
                    false, a0, false, bu, (short)0, accU[0][nt], false, false);
                accU[1][nt] = __builtin_amdgcn_wmma_f32_16x16x4_f32(
                    false, a1, false, bu, (short)0, accU[1][nt], false, false);
            }
        }
        if (s + 1 < NS) {
            const int nxt = cur ^ 1;
#pragma unroll
            for (int it = 0; it < 8; ++it)
                *(float4*)(&lX[nxt][(ldRow + it * 32) * LDSTR + ldC4]) = rx[it];
#pragma unroll
            for (int it = 0; it < 2; ++it) {
                *(float4*)(&lG[nxt][(ldRow + it * 32) * LDSTR + ldC4]) = rg[it];
                *(float4*)(&lU[nxt][(ldRow + it * 32) * LDSTR + ldC4]) = ru[it];
            }
        }
        __syncthreads();
    }

    // Epilogue: C frag layout -> VGPR r: lanes0-15 M=r, lanes16-31 M=r+8; N=lane%16
#pragma unroll
    for (int ms = 0; ms < 2; ++ms) {
        const int mloc0 = mTile + wave * 32 + ms * 16 + ((lane >> 4) << 3);
        float w8[8];
#pragma unroll
        for (int r = 0; r < 8; ++r)
            w8[r] = wrow ? wrow[mBase + mloc0 + r] : 1.0f;
#pragma unroll
        for (int nt = 0; nt < 4; ++nt) {
            const int n = nBase + nt * 16 + lm;
#pragma unroll
            for (int r = 0; r < 8; ++r) {
                float g = accG[ms][nt][r];
                float u = accU[ms][nt][r];
                float sg = g * (1.0f / (1.0f + __expf(-g)));   // silu
                Hdn[(size_t)(mloc0 + r) * I_DIM + n] = sg * u * w8[r];
            }
        }
    }
}

// ---------------------------------------------------------------------------
// GEMM2: y[t,n] (+)= hdn · Wd^T   (K = I_DIM). Same tiling/pipelining.
// ---------------------------------------------------------------------------
__global__ void __launch_bounds__(256)
moe_gemm2_down(const float* __restrict__ Hdn,
               const float* __restrict__ Wd,
               float* __restrict__ Y,
               int mBase, int accumulate) {
    __shared__ float lA[2][256 * LDSTR];
    __shared__ float lW[2][64 * LDSTR];

    const int tid  = threadIdx.x;
    const int wave = tid >> 5;
    const int lane = tid & 31;
    const int lm   = lane & 15;
    const int kh   = (lane >> 4) << 1;

    const int nBase = blockIdx.x * 64;
    const int mTile = blockIdx.y * 256;

    const int ldRow = tid >> 3;
    const int ldC4  = (tid & 7) << 2;

    v8f acc[2][4] = {};

    float4 ra[8], rw[2];
#pragma unroll
    for (int it = 0; it < 8; ++it)
        ra[it] = *(const float4*)(Hdn + (size_t)(mTile + ldRow + it * 32) * I_DIM + ldC4);
#pragma unroll
    for (int it = 0; it < 2; ++it)
        rw[it] = *(const float4*)(Wd + (size_t)(nBase + ldRow + it * 32) * I_DIM + ldC4);
#pragma unroll
    for (int it = 0; it < 8; ++it)
        *(float4*)(&lA[0][(ldRow + it * 32) * LDSTR + ldC4]) = ra[it];
#pragma unroll
    for (int it = 0; it < 2; ++it)
        *(float4*)(&lW[0][(ldRow + it * 32) * LDSTR + ldC4]) = rw[it];
    __syncthreads();

    const int NS = I_DIM / 32;   // 86
    for (int s = 0; s < NS; ++s) {
        const int cur = s & 1;
        if (s + 1 < NS) {
            const int k0 = (s + 1) * 32;
#pragma unroll
            for (int it = 0; it < 8; ++it)
                ra[it] = *(const float4*)(Hdn + (size_t)(mTile + ldRow + it * 32) * I_DIM + k0 + ldC4);
#pragma unroll
            for (int it = 0; it < 2; ++it)
                rw[it] = *(const float4*)(Wd + (size_t)(nBase + ldRow + it * 32) * I_DIM + k0 + ldC4);
        }
        const float* lAr0 = &lA[cur][(wave * 32 + lm) * LDSTR];
        const float* lAr1 = lAr0 + 16 * LDSTR;
#pragma unroll
        for (int kk = 0; kk < 32; kk += 4) {
            v2f a0, a1;
            a0.x = lAr0[kk + kh]; a0.y = lAr0[kk + kh + 1];
            a1.x = lAr1[kk + kh]; a1.y = lAr1[kk + kh + 1];
#pragma unroll
            for (int nt = 0; nt < 4; ++nt) {
                const float* wp = &lW[cur][(nt * 16 + lm) * LDSTR + kk + kh];
                v2f b; b.x = wp[0]; b.y = wp[1];
                acc[0][nt] = __builtin_amdgcn_wmma_f32_16x16x4_f32(
                    false, a0, false, b, (short)0, acc[0][nt], false, false);
                acc[1][nt] = __builtin_amdgcn_wmma_f32_16x16x4_f32(
                    false, a1, false, b, (short)0, acc[1][nt], false, false);
            }
        }
        if (s + 1 < NS) {
            const int nxt = cur ^ 1;
#pragma unroll
            for (int it = 0; it < 8; ++it)
                *(float4*)(&lA[nxt][(ldRow + it * 32) * LDSTR + ldC4]) = ra[it];
#pragma unroll
            for (int it = 0; it < 2; ++it)
                *(float4*)(&lW[nxt][(ldRow + it * 32) * LDSTR + ldC4]) = rw[it];
        }
        __syncthreads();
    }

#pragma unroll
    for (int ms = 0; ms < 2; ++ms) {
        const int mloc0 = mTile + wave * 32 + ms * 16 + ((lane >> 4) << 3);
#pragma unroll
        for (int nt = 0; nt < 4; ++nt) {
            const int n = nBase + nt * 16 + lm;
#pragma unroll
            for (int r = 0; r < 8; ++r) {
                size_t off = (size_t)(mBase + mloc0 + r) * H_DIM + n;
                if (accumulate) Y[off] += acc[ms][nt][r];
                else            Y[off]  = acc[ms][nt][r];
            }
        }
    }
}

// ---------------------------------------------------------------------------
// Launcher: gate -> per token-chunk: shared expert (init y), then 8 routed
// experts accumulate. hdn scratch in d_ws (~22.6 MB chunk).
// ---------------------------------------------------------------------------
extern "C" void kernel_launch(void* const* d_in, const int* in_sizes, int n_in,
                              void* d_out, int out_size, void* d_ws, size_t ws_size,
                              hipStream_t stream) {
    const float* x  = (const float*)d_in[0];
    const float* gw = (const float*)d_in[1];
    const float* Wg = (const float*)d_in[2];
    const float* Wu = (const float*)d_in[3];
    const float* Wd = (const float*)d_in[4];
    const float* Sg = (const float*)d_in[5];
    const float* Su = (const float*)d_in[6];
    const float* Sd = (const float*)d_in[7];
    float* y = (float*)d_out;

    float* w_et = (float*)d_ws;                       // [E, T]
    float* hdn  = w_et + (size_t)E_NUM * T_TOK;       // [MCH, I] scratch

    moe_gate_kernel<<<T_TOK / 8, 256, 0, stream>>>(x, gw, w_et);

    const int MCH = 2048;                             // token chunk
    dim3 blk(256);
    dim3 g1(I_DIM / 64, MCH / 256);                   // 43 x 8
    dim3 g2(H_DIM / 64, MCH / 256);                   // 16 x 8

    for (int mb = 0; mb < T_TOK; mb += MCH) {
        moe_gemm1_swiglu<<<g1, blk, 0, stream>>>(x, Sg, Su, nullptr, hdn, mb);
        moe_gemm2_down <<<g2, blk, 0, stream>>>(hdn, Sd, y, mb, 0);
        for (int e = 0; e < E_NUM; ++e) {
            moe_gemm1_swiglu<<<g1, blk, 0, stream>>>(
                x, Wg + (size_t)e * I_DIM * H_DIM, Wu + (size_t)e * I_DIM * H_DIM,
                w_et + (size_t)e * T_TOK, hdn, mb);
            moe_gemm2_down <<<g2, blk, 0, stream>>>(
                hdn, Wd + (size_t)e * H_DIM * I_DIM, y, mb, 1);
        }
    }
    (void)in_sizes; (void)n_in; (void)out_size; (void)ws_size;
}